// RGCN_90778428768715
// MI455X (gfx1250) — compile-verified
//
#include <hip/hip_runtime.h>
#include <math.h>

// Problem constants hard-coded in the reference module.
#define HD   8     // hidden dim
#define CD   16    // out channels
#define RREL 32    // relations
#define BBAS 15    // bases

typedef float v2f __attribute__((ext_vector_type(2)));
typedef float v8f __attribute__((ext_vector_type(8)));

// ------------------------------------------------------------------
// K0: zero a float region (grid-stride)
// ------------------------------------------------------------------
__global__ void k_zero(float* __restrict__ p, long n) {
  long i = (long)blockIdx.x * blockDim.x + threadIdx.x;
  long stride = (long)gridDim.x * blockDim.x;
  for (; i < n; i += stride) p[i] = 0.0f;
}

// ------------------------------------------------------------------
// K1: counts[t, r] += 1  (per-(target, relation) in-degree)
// ------------------------------------------------------------------
__global__ void k_count(const int* __restrict__ tgt, const int* __restrict__ et,
                        float* __restrict__ counts, int E) {
  int e = blockIdx.x * blockDim.x + threadIdx.x;
  if (e < E) atomicAdd(&counts[(long)tgt[e] * RREL + et[e]], 1.0f);
}

// ------------------------------------------------------------------
// K2: w1[R, NH] = comp1[R, B] @ basis1[B, NH]   via f32 WMMA 16x16x4.
//     One wave per 16-column tile; computes both M-tiles (r=0..15, 16..31)
//     reusing the B fragments. K=15 zero-padded to 16 (4 k-steps).
//     comp1 is staged in LDS zero-padded -> unconditional A loads; the
//     single OOB B element (k=15) is handled with clamp+select (no EXEC
//     branches around loads).
// ------------------------------------------------------------------
__global__ void k_w1_wmma(const float* __restrict__ comp1,   // [R, B] row-major
                          const float* __restrict__ basis1,  // [B, NH]
                          float* __restrict__ w1,            // [R, NH]
                          int NH) {
  __shared__ float compS[RREL * 16];          // comp1 padded K: 15 -> 16
  for (int i = threadIdx.x; i < RREL * 16; i += blockDim.x) {
    int r = i >> 4, k = i & 15;
    compS[i] = (k < BBAS) ? comp1[r * BBAS + k] : 0.0f;
  }
  __syncthreads();                             // all threads reach this

  int lane = threadIdx.x & 31;
  int wave = threadIdx.x >> 5;
  int tile = blockIdx.x * (blockDim.x >> 5) + wave;
  if (tile >= (NH >> 4)) return;               // wave-uniform guard
  int col0 = tile << 4;
  int n    = lane & 15;                        // A-row (m) / B-col (n)
  int khi  = (lane >> 4) << 1;                 // 0 (lanes 0-15) or 2 (16-31)

  v8f acc0 = {0.f,0.f,0.f,0.f,0.f,0.f,0.f,0.f};
  v8f acc1 = {0.f,0.f,0.f,0.f,0.f,0.f,0.f,0.f};

#pragma unroll
  for (int kk = 0; kk < 4; ++kk) {
    int k0  = kk * 4 + khi;                    // <= 14: always in range
    int k1  = k0 + 1;                          // == 15 only for kk=3, hi half
    int k1c = (k1 < BBAS) ? k1 : (BBAS - 1);   // clamped (legal) address
    v2f b, a0, a1;
    b.x = basis1[(long)k0 * NH + col0 + n];
    float by = basis1[(long)k1c * NH + col0 + n];
    b.y = (k1 < BBAS) ? by : 0.0f;             // v_cndmask, no exec branch
    a0.x = compS[n * 16 + k0];
    a0.y = compS[n * 16 + k1];                 // padded slot holds 0
    a1.x = compS[(16 + n) * 16 + k0];
    a1.y = compS[(16 + n) * 16 + k1];
    acc0 = __builtin_amdgcn_wmma_f32_16x16x4_f32(false, a0, false, b,
                                                 (short)0, acc0, false, false);
    acc1 = __builtin_amdgcn_wmma_f32_16x16x4_f32(false, a1, false, b,
                                                 (short)0, acc1, false, false);
  }

  int mhalf = (lane >> 4) << 3;                // D: vgpr i -> row i (+8 hi half)
#pragma unroll
  for (int i = 0; i < 8; ++i) {
    int r = i + mhalf;
    w1[(long)r        * NH + col0 + n] = acc0[i];
    w1[(long)(16 + r) * NH + col0 + n] = acc1[i];
  }
}

// ------------------------------------------------------------------
// K3: layer-1 message + mean scatter:
//     agg1[tgt] += norm * w1[rel, src, :]   (8 floats per edge)
//     w1 (51 MB) is L2-resident (192 MB L2) -> random gathers hit L2.
// ------------------------------------------------------------------
__global__ void k_msg1(const int* __restrict__ src, const int* __restrict__ tgt,
                       const int* __restrict__ et, const float* __restrict__ counts,
                       const float* __restrict__ w1, float* __restrict__ agg1,
                       int E, int NH) {
  int e = blockIdx.x * blockDim.x + threadIdx.x;
  if (e >= E) return;
  int s = src[e], t = tgt[e], r = et[e];
  float norm = 1.0f / fmaxf(counts[(long)t * RREL + r], 1.0f);
  const float* row = &w1[(long)r * NH + (long)s * HD];
  float4 lo = *(const float4*)(row);
  float4 hi = *(const float4*)(row + 4);
  float* dst = &agg1[(long)t * HD];
  atomicAdd(dst + 0, norm * lo.x);
  atomicAdd(dst + 1, norm * lo.y);
  atomicAdd(dst + 2, norm * lo.z);
  atomicAdd(dst + 3, norm * lo.w);
  atomicAdd(dst + 4, norm * hi.x);
  atomicAdd(dst + 5, norm * hi.y);
  atomicAdd(dst + 6, norm * hi.z);
  atomicAdd(dst + 7, norm * hi.w);
}

// ------------------------------------------------------------------
// K4: x = relu(agg1 + root1 + bias1)  (in place)
// ------------------------------------------------------------------
__global__ void k_relu(float* __restrict__ x, const float* __restrict__ root1,
                       const float* __restrict__ bias1, int NH) {
  int i = blockIdx.x * blockDim.x + threadIdx.x;
  if (i < NH) {
    float v = x[i] + root1[i] + bias1[i & (HD - 1)];
    x[i] = v > 0.0f ? v : 0.0f;
  }
}

// ------------------------------------------------------------------
// K5: w2[R*H*C] = comp2 @ basis2  (4096 outputs, trivial)
// ------------------------------------------------------------------
__global__ void k_w2(const float* __restrict__ comp2,
                     const float* __restrict__ basis2,
                     float* __restrict__ w2) {
  int idx = blockIdx.x * blockDim.x + threadIdx.x;
  if (idx < RREL * HD * CD) {
    int r = idx / (HD * CD);
    int i = idx - r * (HD * CD);
    float s = 0.0f;
#pragma unroll
    for (int b = 0; b < BBAS; ++b)
      s += comp2[r * BBAS + b] * basis2[b * HD * CD + i];
    w2[idx] = s;
  }
}

// ------------------------------------------------------------------
// K6: layer-2 message + mean scatter. w2 (16 KB) staged in LDS.
//     agg2[tgt, c] += norm * sum_h x[src, h] * w2[rel, h, c]
// ------------------------------------------------------------------
__global__ void k_msg2(const int* __restrict__ src, const int* __restrict__ tgt,
                       const int* __restrict__ et, const float* __restrict__ counts,
                       const float* __restrict__ x, const float* __restrict__ w2,
                       float* __restrict__ agg2, int E) {
  __shared__ float w2s[RREL * HD * CD];
  for (int i = threadIdx.x; i < RREL * HD * CD; i += blockDim.x) w2s[i] = w2[i];
  __syncthreads();

  int e = blockIdx.x * blockDim.x + threadIdx.x;
  if (e >= E) return;
  int s = src[e], t = tgt[e], r = et[e];
  float norm = 1.0f / fmaxf(counts[(long)t * RREL + r], 1.0f);

  const float4* xr = (const float4*)&x[(long)s * HD];
  float4 lo = xr[0], hi = xr[1];
  float xs[HD] = {lo.x, lo.y, lo.z, lo.w, hi.x, hi.y, hi.z, hi.w};

  const float* wr = &w2s[r * HD * CD];
  float* dst = &agg2[(long)t * CD];
#pragma unroll
  for (int c = 0; c < CD; ++c) {
    float m = 0.0f;
#pragma unroll
    for (int h = 0; h < HD; ++h) m += xs[h] * wr[h * CD + c];
    atomicAdd(dst + c, norm * m);
  }
}

// ------------------------------------------------------------------
// K7: out = log_softmax(agg2 + x @ root2 + bias2) via f32 WMMA.
//     One wave per 16-node tile. D seeded with agg2+bias2; two 16x16x4
//     k-steps cover K=8. Row-wise softmax done on the D fragment with
//     shfl_xor reductions inside each 16-lane half (masks 1,2,4,8).
// ------------------------------------------------------------------
__global__ void k_out(const float* __restrict__ x, const float* __restrict__ root2,
                      const float* __restrict__ agg2, const float* __restrict__ bias2,
                      float* __restrict__ out, int N) {
  int lane = threadIdx.x & 31;
  int wave = threadIdx.x >> 5;
  int tile = blockIdx.x * (blockDim.x >> 5) + wave;
  if (tile >= (N >> 4)) return;           // wave-uniform guard
  int n0    = tile << 4;
  int col   = lane & 15;                  // class index / A-row
  int khi   = (lane >> 4) << 1;
  int mhalf = (lane >> 4) << 3;

  v8f acc;
#pragma unroll
  for (int i = 0; i < 8; ++i)
    acc[i] = agg2[(long)(n0 + i + mhalf) * CD + col] + bias2[col];

#pragma unroll
  for (int kk = 0; kk < 2; ++kk) {
    int k0 = kk * 4 + khi;
    v2f a, b;
    a.x = x[(long)(n0 + col) * HD + k0];
    a.y = x[(long)(n0 + col) * HD + k0 + 1];
    b.x = root2[k0 * CD + col];
    b.y = root2[(k0 + 1) * CD + col];
    acc = __builtin_amdgcn_wmma_f32_16x16x4_f32(false, a, false, b,
                                                (short)0, acc, false, false);
  }

#pragma unroll
  for (int i = 0; i < 8; ++i) {
    float v = acc[i];
    float m = v;
#pragma unroll
    for (int msk = 1; msk < 16; msk <<= 1)
      m = fmaxf(m, __shfl_xor(m, msk, 32));
    float ex = __expf(v - m);
    float ssum = ex;
#pragma unroll
    for (int msk = 1; msk < 16; msk <<= 1)
      ssum += __shfl_xor(ssum, msk, 32);
    out[(long)(n0 + i + mhalf) * CD + col] = v - m - __logf(ssum);
  }
}

// ------------------------------------------------------------------
// Launch
// ------------------------------------------------------------------
extern "C" void kernel_launch(void* const* d_in, const int* in_sizes, int n_in,
                              void* d_out, int out_size, void* d_ws, size_t ws_size,
                              hipStream_t stream) {
  const int*   edge_index = (const int*)d_in[0];   // [2, E]
  const int*   edge_type  = (const int*)d_in[1];   // [E]
  const float* basis1     = (const float*)d_in[2]; // [B, N, H]
  const float* comp1      = (const float*)d_in[3]; // [R, B]
  const float* root1      = (const float*)d_in[4]; // [N, H]
  const float* bias1      = (const float*)d_in[5]; // [H]
  const float* basis2     = (const float*)d_in[6]; // [B, H, C]
  const float* comp2      = (const float*)d_in[7]; // [R, B]
  const float* root2      = (const float*)d_in[8]; // [H, C]
  const float* bias2      = (const float*)d_in[9]; // [C]

  const int E  = in_sizes[1];
  const int NH = in_sizes[4];         // N*H
  const int N  = NH / HD;
  const int* src = edge_index;
  const int* tgt = edge_index + E;

  // Workspace layout (floats): w1 | counts | x(agg1) | w2 | agg2
  // (counts..agg2 contiguous so one zero launch covers all accumulators)
  float* ws     = (float*)d_ws;
  float* w1     = ws;                                   // R*NH (L2-resident)
  float* counts = w1 + (size_t)RREL * NH;               // N*R
  float* xbuf   = counts + (size_t)N * RREL;            // NH
  float* w2     = xbuf + (size_t)NH;                    // R*H*C
  float* agg2   = w2 + (size_t)RREL * HD * CD;          // N*C

  // Zero all accumulators every call (harness does not re-poison between
  // replays). w2 is included (fully overwritten anyway, 16 KB).
  long nzero = (long)N * RREL + (long)NH + (long)RREL * HD * CD + (long)N * CD;
  k_zero<<<1024, 256, 0, stream>>>(counts, nzero);

  k_count<<<(E + 255) / 256, 256, 0, stream>>>(tgt, edge_type, counts, E);

  int tiles1 = NH >> 4;                                 // 25000 column tiles
  k_w1_wmma<<<(tiles1 + 7) / 8, 256, 0, stream>>>(comp1, basis1, w1, NH);

  k_msg1<<<(E + 255) / 256, 256, 0, stream>>>(src, tgt, edge_type, counts,
                                              w1, xbuf, E, NH);

  k_relu<<<(NH + 255) / 256, 256, 0, stream>>>(xbuf, root1, bias1, NH);

  k_w2<<<(RREL * HD * CD + 255) / 256, 256, 0, stream>>>(comp2, basis2, w2);

  k_msg2<<<(E + 255) / 256, 256, 0, stream>>>(src, tgt, edge_type, counts,
                                              xbuf, w2, agg2, E);

  int tiles2 = N >> 4;                                  // 3125 node tiles
  k_out<<<(tiles2 + 7) / 8, 256, 0, stream>>>(xbuf, root2, agg2, bias2,
                                              (float*)d_out, N);
}